// Gridup_23261542875221
// MI455X (gfx1250) — compile-verified
//
#include <hip/hip_runtime.h>
#include <hip/hip_bf16.h>

typedef __attribute__((ext_vector_type(2))) float v2f;
typedef __attribute__((ext_vector_type(8))) float v8f;

// Problem dims (fixed by setup_inputs):
//   grid:     [N=2, K=32, C=32, h=64, w=64]  fp32
//   guidemap: [N=2, C=32, H=1024, W=1024]    fp32
//   out:      [N=2, K=32, H=1024, W=1024]    fp32
// out[n,k,y,x] = sum_c grid[n,k,c,y/16,x/16] * guidemap[n,c,y,x]
#define GN 2
#define GK 32
#define GC 32
#define Gh 64
#define Gw 64
#define IH 1024
#define IW 1024
#define PLANE_SHIFT 20      // floats per channel plane = IH*IW = 1<<20
#define LDS_STRIDE 34       // even (8B-aligned v2f reads), conflict-free

__global__ __launch_bounds__(256) void gridup_wmma_kernel(
    const float* __restrict__ grid,
    const float* __restrict__ guide,
    float* __restrict__ out)
{
    __shared__ float Msh[GK * LDS_STRIDE];

    // One workgroup per grid cell: cells = N * h * w = 2*64*64 = 8192
    const int cell = blockIdx.x;
    const int n  = cell >> 12;         // / (64*64)
    const int by = (cell >> 6) & 63;   // cell row
    const int bx = cell & 63;          // cell col

    const int tid  = threadIdx.x;
    const int lane = tid & 31;
    const int wv   = tid >> 5;         // 0..7
    const int ktile  = wv & 1;         // which 16 output channels
    const int pgroup = wv >> 1;        // which 64 pixels of the 256-pixel tile

    // ---- Cooperative load of the 32x32 cell matrix M into LDS ----
    // grid[n, ko, c, by, bx]; element stride over (ko,c) is h*w = 4096 floats.
    // Uniform SGPR base + 32-bit VGPR offsets (max ~16.7MB, fits easily).
    const float* gcell = grid + (size_t)n * (GK * GC * Gh * Gw)
                              + (unsigned)(by * Gw + bx);
    #pragma unroll
    for (int i = 0; i < 4; ++i) {
        unsigned e  = (unsigned)tid + i * 256u;   // 0..1023
        unsigned ko = e >> 5;
        unsigned c  = e & 31u;
        Msh[ko * LDS_STRIDE + c] = gcell[e << 12];  // e * h*w
    }
    __syncthreads();

    const int lo = lane & 15;          // pixel / row selector within half-wave
    const int hi = lane >> 4;          // half-wave selector

    // ---- A fragments (16x4 chunks of M) in WMMA layout from LDS ----
    // lanes 0-15: row M=lane, K = kc*4 + {0,1}; lanes 16-31: K = kc*4 + {2,3}
    v2f a[8];
    {
        const int row = ktile * 16 + lo;
        #pragma unroll
        for (int kc = 0; kc < 8; ++kc) {
            const int c0 = kc * 4 + 2 * hi;   // even -> 8B aligned
            a[kc] = *(const v2f*)(&Msh[row * LDS_STRIDE + c0]);
        }
    }

    // Uniform bases (stay in SGPRs); all per-lane offsets are 32-bit floats
    // indices (max 128MB byte span << 2^31) so loads/stores use saddr+voffset.
    const float* gbase = guide + ((size_t)n << (PLANE_SHIFT + 5));  // n*C*plane
    float*       obase = out   + ((size_t)n << (PLANE_SHIFT + 5));  // n*K*plane

    const unsigned y0 = (unsigned)(by * 16);
    const unsigned x  = (unsigned)(bx * 16 + lo);
    const unsigned kout0 = (unsigned)(ktile * 16 + 8 * hi);  // D-tile channel base
    const unsigned chi2  = (unsigned)(2 * hi);               // B-fragment channel bias

    // ---- 4 output tiles per wave: 16 channels x 16 pixels each ----
    #pragma unroll
    for (int pt = 0; pt < 4; ++pt) {
        const unsigned px = (unsigned)(pgroup * 64 + pt * 16); // 16-aligned base
        const unsigned y  = y0 + (px >> 4);
        const unsigned rowoff = y * IW + x;                    // < 2^20

        v8f acc = {};
        #pragma unroll
        for (int kc = 0; kc < 8; ++kc) {
            // B fragment (4x16): lanes 0-15 hold channels c0,c0+1 for pixel=lo,
            // lanes 16-31 hold channels c0+2,c0+3 for the same 16 pixels.
            const unsigned c0 = (unsigned)(kc * 4) + chi2;
            const unsigned bidx = (c0 << PLANE_SHIFT) + rowoff;
            v2f b;
            b.x = gbase[bidx];                        // channel c0
            b.y = gbase[bidx + (1u << PLANE_SHIFT)];  // c0+1: +4MB fits imm24
            acc = __builtin_amdgcn_wmma_f32_16x16x4_f32(
                /*neg_a=*/false, a[kc], /*neg_b=*/false, b,
                /*c_mod=*/(short)0, acc, /*reuse_a=*/false, /*reuse_b=*/false);
        }

        // ---- Store D tile. C/D layout: VGPR v holds channel (kout0+v),
        //      lane lo = pixel. 64B contiguous per half-wave per store. ----
        const unsigned oidx = (kout0 << PLANE_SHIFT) + rowoff;
        #pragma unroll
        for (int v = 0; v < 8; ++v) {
            obase[oidx + ((unsigned)v << PLANE_SHIFT)] = acc[v];
        }
    }
}

extern "C" void kernel_launch(void* const* d_in, const int* in_sizes, int n_in,
                              void* d_out, int out_size, void* d_ws, size_t ws_size,
                              hipStream_t stream) {
    const float* grid_p  = (const float*)d_in[0];
    const float* guide_p = (const float*)d_in[1];
    float* out_p = (float*)d_out;

    const int cells = GN * Gh * Gw;   // 8192 workgroups, 256 threads (8 waves)
    hipLaunchKernelGGL(gridup_wmma_kernel, dim3(cells), dim3(256), 0, stream,
                       grid_p, guide_p, out_p);
}